// SutraV04_70325794505397
// MI455X (gfx1250) — compile-verified
//
#include <hip/hip_runtime.h>
#include <math.h>

// ============================================================================
// MI455X (gfx1250) forward pass for the hierarchical GRU + message-passing +
// sparse-retrieval model. All dense GEMMs run on the matrix pipe via
// V_WMMA_F32_16X16X4_F32 (fp32 in / fp32 accum) to match the fp32 reference's
// precision class while using CDNA5 WMMA hardware.
// GEMM: 128x64 block tile, 8 wave32 waves, each wave owns a 32x32 C slab
// (2x2 register blocking, 4 WMMAs per k-step). Global->LDS staging is b128-
// vectorized and double-buffered (register prefetch), one barrier per chunk.
// Algebraic simplifications: message MLP first layer split into shared-self +
// gathered-neighbor halves; neighbor mean commuted before msg_w2; upd MLP
// first layer split into two accumulating GEMMs; o_w projection accumulates
// directly into `output` to form `combined`.
// ============================================================================

typedef __attribute__((ext_vector_type(2))) float v2f;
typedef __attribute__((ext_vector_type(8))) float v8f;

#define BB   32
#define TT   2048
#define PSZ  4
#define NNP  512            // patches per batch (N)
#define DD   256
#define BN   (BB*NNP)       // 16384
#define MTOK (BB*TT)        // 65536 token rows
#define VV   256
#define WINW 4
#define KRET 8

__device__ __forceinline__ float geluf(float x){ return 0.5f*x*(1.0f+erff(x*0.70710678118654752440f)); }
__device__ __forceinline__ float sigmf(float x){ return 1.0f/(1.0f+expf(-x)); }

// ---------------------------------------------------------------- fill
__global__ void k_fill(float* p, float v, long n){
  long i = (long)blockIdx.x*blockDim.x + threadIdx.x;
  if (i < n) p[i] = v;
}

// ------------------------------------------------- transpose (R x C -> C x R)
__global__ void k_transpose(const float* __restrict__ in, float* __restrict__ out, int R, int C){
  long i = (long)blockIdx.x*blockDim.x + threadIdx.x;
  if (i < (long)R*C){
    int r = (int)(i / C), c = (int)(i % C);
    out[(long)c*R + r] = in[i];
  }
}

// ---------------------------------------------------------------- embedding
// hin[row, d] = emb[x[row], d] + pos[row % 4, d]   (row = flat token index)
__global__ void k_embed(const int* __restrict__ x, const float* __restrict__ emb,
                        const float* __restrict__ pos, float* __restrict__ hin, long n){
  long i = (long)blockIdx.x*blockDim.x + threadIdx.x;
  if (i < n){
    long row = i >> 8; int d = (int)(i & 255);
    int tok = x[row]; int p = (int)(row & 3);
    hin[i] = emb[(long)tok*DD + d] + pos[p*DD + d];
  }
}

// ============================================================================
// WMMA f32 GEMM:  C = act( alpha * (A @ B + bias + (accum ? C : 0)) )
//   A: M x K (row stride lda); B: K x N row-major (ldb), or if transB,
//   element (k,n) read from Bm[n*ldb + k].
//   Requirements: M % 128 == 0, N % 64 == 0, K % 16 == 0 (all call sites obey).
// ============================================================================
__global__ void __launch_bounds__(256) k_gemm(
    const float* __restrict__ A, const float* __restrict__ Bm,
    const float* __restrict__ bias, float* __restrict__ C,
    int M, int N, int K, int lda, int ldb, int ldc,
    long sA, long sB, long sC,
    int transB, int accum, int act, float alpha)
{
  A  += (long)blockIdx.z * sA;
  Bm += (long)blockIdx.z * sB;
  C  += (long)blockIdx.z * sC;

  // double-buffered tiles; row strides padded (20 / 68 floats) so b128 LDS
  // ops stay 16B-aligned and fragment reads are bank-conflict-free.
  __shared__ __align__(16) float As[2][128][20];   // [buf][m][k]
  __shared__ __align__(16) float Bs[2][16][68];    // [buf][k][n]

  const int tid  = threadIdx.x;
  const int lane = tid & 31;
  const int wave = tid >> 5;          // 0..7
  const int half = lane >> 4;         // 0|1
  const int lr   = lane & 15;
  const int wm   = wave & 3;          // 4 waves along M (32 rows each)
  const int wn   = wave >> 2;         // 2 waves along N (32 cols each)
  const int m0   = blockIdx.x * 128;
  const int n0   = blockIdx.y * 64;

  // ---- staging assignments (hoisted pointers, advanced by constant strides)
  const int am0 = tid >> 2, ac0 = (tid & 3) * 4;   // A rows 0..63
  const int am1 = am0 + 64, ac1 = ac0;             // A rows 64..127
  const float* Ap0 = A + (long)(m0 + am0)*lda + ac0;
  const float* Ap1 = A + (long)(m0 + am1)*lda + ac1;
  int bk, bn; const float* Bp; long stepB;
  if (transB){ bn = tid >> 2;  bk = (tid & 3)  * 4; Bp = Bm + (long)(n0 + bn)*ldb + bk; stepB = 16; }
  else       { bk = tid >> 4;  bn = (tid & 15) * 4; Bp = Bm + (long)bk*ldb + (n0 + bn); stepB = (long)16*ldb; }

  union V8 { v8f v; float f[8]; };
  V8 acc[2][2];
  const int colJ0 = n0 + wn*32 + lr;
  const int colJ1 = colJ0 + 16;
  const float bb0 = bias ? bias[colJ0] : 0.0f;
  const float bb1 = bias ? bias[colJ1] : 0.0f;
#pragma unroll
  for (int i = 0; i < 2; ++i){
#pragma unroll
    for (int r = 0; r < 8; ++r){
      int row = m0 + wm*32 + i*16 + half*8 + r;
      float a0 = bb0, a1 = bb1;
      if (accum){ a0 += C[(long)row*ldc + colJ0]; a1 += C[(long)row*ldc + colJ1]; }
      acc[i][0].f[r] = a0; acc[i][1].f[r] = a1;
    }
  }

  // ---- prefetch chunk 0 into registers
  float4 aR0 = *(const float4*)Ap0; Ap0 += 16;
  float4 aR1 = *(const float4*)Ap1; Ap1 += 16;
  float4 bR  = *(const float4*)Bp;  Bp  += stepB;

  const int kIters = K >> 4;
  for (int it = 0; it < kIters; ++it){
    const int buf = it & 1;
    *(float4*)&As[buf][am0][ac0] = aR0;
    *(float4*)&As[buf][am1][ac1] = aR1;
    if (transB){
      Bs[buf][bk+0][bn] = bR.x; Bs[buf][bk+1][bn] = bR.y;
      Bs[buf][bk+2][bn] = bR.z; Bs[buf][bk+3][bn] = bR.w;
    } else {
      *(float4*)&Bs[buf][bk][bn] = bR;
    }
    __syncthreads();                       // single barrier per chunk
    if (it + 1 < kIters){                  // prefetch next chunk (uniform branch)
      aR0 = *(const float4*)Ap0; Ap0 += 16;
      aR1 = *(const float4*)Ap1; Ap1 += 16;
      bR  = *(const float4*)Bp;  Bp  += stepB;
    }
#pragma unroll
    for (int kk = 0; kk < 4; ++kk){
      // ISA layouts: A 16x4 -> lane half selects K {0,1} vs {2,3};
      // B 4x16 -> VGPR r holds K=r (lanes 0-15) and K=r+2 (lanes 16-31).
      const int kb = kk*4 + half*2;
      v2f a0, a1, b0, b1;
      a0.x = As[buf][wm*32      + lr][kb]; a0.y = As[buf][wm*32      + lr][kb+1];
      a1.x = As[buf][wm*32 + 16 + lr][kb]; a1.y = As[buf][wm*32 + 16 + lr][kb+1];
      b0.x = Bs[buf][kb][wn*32 + lr];      b0.y = Bs[buf][kb+1][wn*32 + lr];
      b1.x = Bs[buf][kb][wn*32 + 16 + lr]; b1.y = Bs[buf][kb+1][wn*32 + 16 + lr];
      acc[0][0].v = __builtin_amdgcn_wmma_f32_16x16x4_f32(false, a0, false, b0, (short)0, acc[0][0].v, false, false);
      acc[0][1].v = __builtin_amdgcn_wmma_f32_16x16x4_f32(false, a0, false, b1, (short)0, acc[0][1].v, false, false);
      acc[1][0].v = __builtin_amdgcn_wmma_f32_16x16x4_f32(false, a1, false, b0, (short)0, acc[1][0].v, false, false);
      acc[1][1].v = __builtin_amdgcn_wmma_f32_16x16x4_f32(false, a1, false, b1, (short)0, acc[1][1].v, false, false);
    }
  }

#pragma unroll
  for (int i = 0; i < 2; ++i){
#pragma unroll
    for (int r = 0; r < 8; ++r){
      int row = m0 + wm*32 + i*16 + half*8 + r;
      float v0 = alpha * acc[i][0].f[r];
      float v1 = alpha * acc[i][1].f[r];
      if (act == 1){ v0 = geluf(v0); v1 = geluf(v1); }
      C[(long)row*ldc + colJ0] = v0;
      C[(long)row*ldc + colJ1] = v1;
    }
  }
}

// ---------------------------------------------------------------- GRU gates
__global__ void k_gru_gate(const float* __restrict__ X1, const float* __restrict__ X2,
                           float* __restrict__ h, float* __restrict__ seq, int p, long n){
  long i = (long)blockIdx.x*blockDim.x + threadIdx.x;
  if (i >= n) return;
  long row = i >> 8; int d = (int)(i & 255);
  long b3 = row*768 + d;
  float ir = X1[b3], iz = X1[b3+256], inn = X1[b3+512];
  float hr = X2[b3], hz = X2[b3+256], hn  = X2[b3+512];
  float hv = h[i];
  float r  = sigmf(ir + hr);
  float z  = sigmf(iz + hz);
  float nn = tanhf(inn + r*hn);
  float ho = (1.0f - z)*nn + z*hv;
  h[i] = ho;
  seq[(row*4 + (long)p)*256 + d] = ho;
}

// --------------------------------------------- LayerNorm (D=256, 1 block/row)
// out[row] = LN( x[row] + (res ? res[row/resDiv] : 0) ) * g + b
__global__ void __launch_bounds__(256) k_ln(const float* __restrict__ x,
                                            const float* __restrict__ res, int resDiv,
                                            const float* __restrict__ g, const float* __restrict__ b,
                                            float* __restrict__ out){
  __shared__ float red[256];
  long row = blockIdx.x; int d = threadIdx.x;
  float v = x[row*256 + d];
  if (res) v += res[(row / resDiv)*256 + d];
  red[d] = v; __syncthreads();
  for (int s = 128; s > 0; s >>= 1){ if (d < s) red[d] += red[d+s]; __syncthreads(); }
  float mean = red[0] * (1.0f/256.0f); __syncthreads();
  float dv = v - mean;
  red[d] = dv*dv; __syncthreads();
  for (int s = 128; s > 0; s >>= 1){ if (d < s) red[d] += red[d+s]; __syncthreads(); }
  float var = red[0] * (1.0f/256.0f);
  out[row*256 + d] = dv * rsqrtf(var + 1e-5f) * g[d] + b[d];
}

// ------------------------------------------------------- mean over PS=4 rows
__global__ void k_meanP(const float* __restrict__ local, float* __restrict__ out, long n){
  long i = (long)blockIdx.x*blockDim.x + threadIdx.x;
  if (i >= n) return;
  long row = i >> 8; int d = (int)(i & 255);
  long base = row*1024 + d;
  out[i] = 0.25f*(local[base] + local[base+256] + local[base+512] + local[base+768]);
}

// ---- neighbor gather: m1mean[b,n,d] = mean_j gelu(hA[b,n,d] + hB[b,idx,d])
// hA already contains msg_b1.
__global__ void k_msg(const float* __restrict__ hA, const float* __restrict__ hB,
                      float* __restrict__ out, long n){
  long i = (long)blockIdx.x*blockDim.x + threadIdx.x;
  if (i >= n) return;
  long bn = i >> 8; int d = (int)(i & 255);
  int  nn = (int)(bn % NNP);
  long b  = bn / NNP;
  float av = hA[i];
  float s = 0.0f; int cnt = 0;
#pragma unroll
  for (int j = 0; j <= WINW; ++j){
    int idx = nn - WINW + j;
    if (idx >= 0){ s += geluf(av + hB[((long)b*NNP + idx)*256 + d]); ++cnt; }
  }
  out[i] = s / (float)cnt;
}

// ------------------------------------------- halting probability per batch b
__global__ void __launch_bounds__(256) k_halt(const float* __restrict__ h,
                                              const float* __restrict__ hw,
                                              const float* __restrict__ hb,
                                              float* __restrict__ hp){
  __shared__ float red[256];
  int b = blockIdx.x, d = threadIdx.x;
  const float* base = h + ((long)b*NNP)*256 + d;
  float s = 0.0f;
  for (int n = 0; n < NNP; ++n) s += base[(long)n*256];
  red[d] = (s * (1.0f/(float)NNP)) * hw[d];
  __syncthreads();
  for (int st = 128; st > 0; st >>= 1){ if (d < st) red[d] += red[d+st]; __syncthreads(); }
  if (d == 0) hp[b] = sigmf(red[0] + hb[0]);
}

__global__ void k_scale_out(const float* __restrict__ h, const float* __restrict__ hp,
                            float* __restrict__ out, long n){
  long i = (long)blockIdx.x*blockDim.x + threadIdx.x;
  if (i >= n) return;
  long b = i / ((long)NNP*DD);
  out[i] = hp[b] * h[i];
}
__global__ void k_add_rem(const float* __restrict__ h, const float* __restrict__ hp,
                          float* __restrict__ out, long n){
  long i = (long)blockIdx.x*blockDim.x + threadIdx.x;
  if (i >= n) return;
  long b = i / ((long)NNP*DD);
  out[i] += (1.0f - hp[b]) * h[i];
}

// ---------------------------------------------------------------- KL scalar
// steps 0,1: hp_mean=0; step 2: mean(hp); step 3: hp_mean=1.
__global__ void k_kl(const float* __restrict__ hp, float* __restrict__ out){
  if (threadIdx.x == 0 && blockIdx.x == 0){
    float m = 0.0f;
    for (int b = 0; b < BB; ++b) m += hp[b];
    m *= (1.0f/(float)BB);
    const float L = 0.2f;
    float hv[4] = {0.0f, 0.0f, m, 1.0f};
    float kl = 0.0f, p = L;
    for (int s = 0; s < 4; ++s){
      kl += p * (logf(p) - logf(hv[s] + 1e-8f));
      p *= (1.0f - L);
    }
    out[0] = kl * 0.25f;
  }
}

// ------------------- top-8 causal + softmax + weighted gather of v (1 blk/row)
__global__ void __launch_bounds__(256) k_topk_attn(const float* __restrict__ scores,
                                                   const float* __restrict__ v,
                                                   float* __restrict__ retr){
  __shared__ float sval[256]; __shared__ int sidx[256];
  __shared__ int   selI[KRET]; __shared__ float selV[KRET]; __shared__ float w[KRET];
  const int n = blockIdx.x, b = blockIdx.y, tid = threadIdx.x;
  const float* row = scores + ((long)b*NNP + n)*NNP;

  for (int it = 0; it < KRET; ++it){
    float best = -3.4e38f; int bi = -1;
    for (int m = tid; m <= n; m += 256){
      bool used = false;
      for (int j = 0; j < it; ++j) if (selI[j] == m) used = true;
      if (!used){
        float val = row[m];
        if (val > best || (val == best && (bi < 0 || m < bi))){ best = val; bi = m; }
      }
    }
    sval[tid] = best; sidx[tid] = bi; __syncthreads();
    for (int s = 128; s > 0; s >>= 1){
      if (tid < s){
        float v2 = sval[tid+s]; int i2 = sidx[tid+s];
        if (i2 >= 0 && (sidx[tid] < 0 || v2 > sval[tid] ||
                        (v2 == sval[tid] && i2 < sidx[tid]))){
          sval[tid] = v2; sidx[tid] = i2;
        }
      }
      __syncthreads();
    }
    if (tid == 0){ selI[it] = sidx[0]; selV[it] = sval[0]; }
    __syncthreads();
  }

  if (tid == 0){
    float mx = -3.4e38f;
    for (int i = 0; i < KRET; ++i) if (selI[i] >= 0 && selV[i] > mx) mx = selV[i];
    float ssum = 0.0f;
    for (int i = 0; i < KRET; ++i){
      float e = (selI[i] >= 0) ? expf(selV[i] - mx) : 0.0f;
      w[i] = e; ssum += e;
    }
    float inv = 1.0f/ssum;
    for (int i = 0; i < KRET; ++i) w[i] *= inv;
  }
  __syncthreads();

  float r = 0.0f;
  for (int i = 0; i < KRET; ++i){
    int mi = selI[i];
    if (mi >= 0) r += w[i] * v[((long)b*NNP + mi)*256 + tid];
  }
  retr[((long)b*NNP + n)*256 + tid] = r;
}

// ============================================================================
extern "C" void kernel_launch(void* const* d_in, const int* in_sizes, int n_in,
                              void* d_out, int out_size, void* d_ws, size_t ws_size,
                              hipStream_t stream)
{
  (void)in_sizes; (void)n_in; (void)out_size;
  const int*   x        = (const int*)  d_in[0];
  const float* emb      = (const float*)d_in[1];
  const float* pos      = (const float*)d_in[2];
  const float* gru_wih  = (const float*)d_in[3];
  const float* gru_whh  = (const float*)d_in[4];
  const float* gru_bih  = (const float*)d_in[5];
  const float* gru_bhh  = (const float*)d_in[6];
  const float* pp_ln_g  = (const float*)d_in[7];
  const float* pp_ln_b  = (const float*)d_in[8];
  const float* sum_w    = (const float*)d_in[9];
  const float* sum_b    = (const float*)d_in[10];
  const float* msg_w1   = (const float*)d_in[11];
  const float* msg_b1   = (const float*)d_in[12];
  const float* msg_w2   = (const float*)d_in[13];
  const float* msg_b2   = (const float*)d_in[14];
  const float* upd_w1   = (const float*)d_in[15];
  const float* upd_b1   = (const float*)d_in[16];
  const float* upd_w2   = (const float*)d_in[17];
  const float* upd_b2   = (const float*)d_in[18];
  const float* halt_w   = (const float*)d_in[19];
  const float* halt_b   = (const float*)d_in[20];
  const float* mp_ln_g  = (const float*)d_in[21];
  const float* mp_ln_b  = (const float*)d_in[22];
  const float* q_w      = (const float*)d_in[23];
  const float* q_b      = (const float*)d_in[24];
  const float* k_w      = (const float*)d_in[25];
  const float* k_b      = (const float*)d_in[26];
  const float* v_w      = (const float*)d_in[27];
  const float* v_b      = (const float*)d_in[28];
  const float* o_w      = (const float*)d_in[29];
  const float* o_b      = (const float*)d_in[30];
  const float* br_w     = (const float*)d_in[31];
  const float* br_b     = (const float*)d_in[32];
  const float* f_ln_g   = (const float*)d_in[33];
  const float* f_ln_b   = (const float*)d_in[34];
  const float* head_w   = (const float*)d_in[35];

  float* logits = (float*)d_out;                       // [MTOK, VV]
  float* klout  = logits + (size_t)MTOK*VV;            // scalar

  // ---------------- workspace layout (floats), ~371 MB total ----------------
  float* ws = (float*)d_ws;
  size_t off = 0;
  auto alloc = [&](size_t nf){ float* p = ws + off; off += nf; return p; };
  float* WihT = alloc((size_t)768*256);
  float* WhhT = alloc((size_t)768*256);
  float* hin  = alloc((size_t)MTOK*DD);    // -> reused as `local`
  float* seq  = alloc((size_t)MTOK*DD);    // -> reused as `scores`, then `ln_final`
  float* X1   = alloc((size_t)BN*768);
  float* X2   = alloc((size_t)BN*768);
  float* h    = alloc((size_t)BN*DD);
  float* pA   = alloc((size_t)BN*DD);      // hA      / q
  float* pB   = alloc((size_t)BN*DD);      // hB      / k
  float* pC   = alloc((size_t)BN*DD);      // m1mean  / v
  float* pD   = alloc((size_t)BN*DD);      // msgs    / retrieved
  float* pE   = alloc((size_t)BN*DD);      // t1      / broad
  float* pF   = alloc((size_t)BN*DD);      // u2
  float* outb = alloc((size_t)BN*DD);      // halting output -> combined
  float* hp   = alloc(64);
  if (ws_size < off * sizeof(float)) return;   // insufficient scratch

  auto gemm = [&](const float* A, const float* Bm, const float* bias, float* C,
                  int M, int N, int K, int lda, int ldb, int ldc,
                  long sA, long sB, long sC, int batch,
                  int transB, int accum, int act, float alpha){
    dim3 g(M/128, N/64, batch), blk(256);
    k_gemm<<<g, blk, 0, stream>>>(A, Bm, bias, C, M, N, K, lda, ldb, ldc,
                                  sA, sB, sC, transB, accum, act, alpha);
  };
  const long nBND  = (long)BN*DD;     // 4,194,304
  const long nTOKD = (long)MTOK*DD;   // 16,777,216
  const int  EB = 256;

  // ---- weight transposes: (768,256) -> (256,768) for x @ W.T
  k_transpose<<<(768*256)/EB, EB, 0, stream>>>(gru_wih, WihT, 768, 256);
  k_transpose<<<(768*256)/EB, EB, 0, stream>>>(gru_whh, WhhT, 768, 256);

  // ---- embedding + positional
  k_embed<<<(int)(nTOKD/EB), EB, 0, stream>>>(x, emb, pos, hin, nTOKD);

  // ---- GRU over PS=4 steps (h starts at 0)
  k_fill<<<(int)(nBND/EB), EB, 0, stream>>>(h, 0.0f, nBND);
  for (int p = 0; p < PSZ; ++p){
    gemm(hin + p*DD, WihT, gru_bih, X1, BN, 768, 256, PSZ*DD, 768, 768, 0,0,0,1, 0,0,0, 1.0f);
    gemm(h,          WhhT, gru_bhh, X2, BN, 768, 256, 256,    768, 768, 0,0,0,1, 0,0,0, 1.0f);
    k_gru_gate<<<(int)(nBND/EB), EB, 0, stream>>>(X1, X2, h, seq, p, nBND);
  }

  // ---- local = LN(seq)  (written over hin region)
  float* local = hin;
  k_ln<<<MTOK, 256, 0, stream>>>(seq, nullptr, 1, pp_ln_g, pp_ln_b, local);

  // ---- summaries: h = mean_PS(local) @ sum_w + sum_b
  k_meanP<<<(int)(nBND/EB), EB, 0, stream>>>(local, pA, nBND);
  gemm(pA, sum_w, sum_b, h, BN, 256, 256, 256, 256, 256, 0,0,0,1, 0,0,0, 1.0f);

  // ---- 4 rounds of windowed message passing with halting
  for (int s = 0; s < 4; ++s){
    gemm(h, msg_w1,              msg_b1,  pA, BN,256,256, 256,256,256, 0,0,0,1, 0,0,0, 1.0f); // self half (+b1)
    gemm(h, msg_w1 + 256*256,    nullptr, pB, BN,256,256, 256,256,256, 0,0,0,1, 0,0,0, 1.0f); // neighbor half
    k_msg<<<(int)(nBND/EB), EB, 0, stream>>>(pA, pB, pC, nBND);                                // masked mean of gelu
    gemm(pC, msg_w2, msg_b2, pD, BN,256,256, 256,256,256, 0,0,0,1, 0,0,0, 1.0f);               // msgs
    gemm(h,  upd_w1,           upd_b1,  pE, BN,256,256, 256,256,256, 0,0,0,1, 0,0,0, 1.0f);    // t1  = h@U1a + b1
    gemm(pD, upd_w1 + 256*256, nullptr, pE, BN,256,256, 256,256,256, 0,0,0,1, 0,1,1, 1.0f);    // t1  = gelu(t1 + msgs@U1b)
    gemm(pE, upd_w2, upd_b2, pF, BN,256,256, 256,256,256, 0,0,0,1, 0,0,0, 1.0f);               // u2
    k_ln<<<BN, 256, 0, stream>>>(h, pF, 1, mp_ln_g, mp_ln_b, h);                               // h = LN(h + u2)
    if (s == 2){
      k_halt<<<BB, 256, 0, stream>>>(h, halt_w, halt_b, hp);
      k_scale_out<<<(int)(nBND/EB), EB, 0, stream>>>(h, hp, outb, nBND);     // output  = hp * h2
    } else if (s == 3){
      k_add_rem<<<(int)(nBND/EB), EB, 0, stream>>>(h, hp, outb, nBND);       // output += (1-hp) * h3
    }
  }
  k_kl<<<1, 32, 0, stream>>>(hp, klout);

  // ---- sparse retrieval attention
  float* q = pA; float* kk = pB; float* vv = pC; float* retr = pD; float* broad = pE;
  gemm(outb, q_w, q_b, q,  BN,256,256, 256,256,256, 0,0,0,1, 0,0,0, 1.0f);
  gemm(outb, k_w, k_b, kk, BN,256,256, 256,256,256, 0,0,0,1, 0,0,0, 1.0f);
  gemm(outb, v_w, v_b, vv, BN,256,256, 256,256,256, 0,0,0,1, 0,0,0, 1.0f);
  float* scores = seq;   // seq dead; 32 * 512*512 floats
  gemm(q, kk, nullptr, scores, NNP, NNP, 256, 256, 256, NNP,
       (long)NNP*DD, (long)NNP*DD, (long)NNP*NNP, BB, 1, 0, 0, 0.0625f);     // q@k^T / sqrt(D)
  dim3 tg(NNP, BB);
  k_topk_attn<<<tg, 256, 0, stream>>>(scores, vv, retr);
  gemm(retr, o_w, o_b, outb, BN,256,256, 256,256,256, 0,0,0,1, 0,1,0, 1.0f); // combined = output + retr@o_w + o_b
  gemm(outb, br_w, br_b, broad, BN,256,256, 256,256,256, 0,0,0,1, 0,0,0, 1.0f);

  // ---- final = LN(local + broadcast(broad)); logits = final @ head_w
  float* lnf = seq;      // scores dead
  k_ln<<<MTOK, 256, 0, stream>>>(local, broad, PSZ, f_ln_g, f_ln_b, lnf);
  gemm(lnf, head_w, nullptr, logits, MTOK, VV, 256, 256, 256, 256, 0,0,0,1, 0,0,0, 1.0f);
}